// Leaf2Embedding_56607668961387
// MI455X (gfx1250) — compile-verified
//
#include <hip/hip_runtime.h>
#include <stdint.h>

// Problem constants (from the reference)
#define N_SPLIT      100
#define MAXLEAF      64
#define EMBSIZE      32
#define N_TREES      2000
#define M_PER_SPLIT  20                      // trees per split
#define BATCH        4096
#define MROWS        (M_PER_SPLIT * MAXLEAF) // 1280 rows per split table
#define TBL_FLOATS   (MROWS * EMBSIZE)       // 40960 floats
#define TBL_BYTES    (TBL_FLOATS * 4)        // 163840 B = 160 KB (half a WGP's LDS)

#define THREADS      256
#define CHUNK        128                     // batches per block

__global__ __launch_bounds__(THREADS)
void Leaf2Embedding_gather_kernel(const int* __restrict__ leaves,
                                  const float* __restrict__ embed_w,
                                  float* __restrict__ out) {
    __shared__ float tbl[TBL_FLOATS];        // 160 KB: this split's embedding table

    const int s    = blockIdx.x;             // split id 0..99
    const int base = blockIdx.y * CHUNK;     // first batch row for this block
    const int tid  = threadIdx.x;

    // ---- Stage the split's table into LDS with gfx1250 async-to-LDS DMA ----
    // GVS addressing: mem = SGPR64(base) + VGPR32(byte offset); LDS dst addr in VGPR.
    {
        const float* src = embed_w + (size_t)s * TBL_FLOATS;
        // low 32 bits of the generic (flat) pointer == LDS byte offset (aperture map)
        unsigned lds_base = (unsigned)(uintptr_t)&tbl[0];
#pragma unroll 4
        for (int p = 0; p < TBL_BYTES / (THREADS * 16); ++p) {   // 40 passes of b128
            unsigned off      = (unsigned)(p * THREADS + tid) * 16u;
            unsigned lds_addr = lds_base + off;
            asm volatile("global_load_async_to_lds_b128 %0, %1, %2"
                         :
                         : "v"(lds_addr), "v"(off), "s"(src)
                         : "memory");
        }
        asm volatile("s_wait_asynccnt 0" ::: "memory");          // per-wave ASYNCcnt drain
    }
    __syncthreads();                                             // make LDS visible to all waves

    // ---- Gather + sum ----
    // Lane mapping: 8 lanes (q=0..7) hold one 32-float output row as float4 each.
    // A wave32 therefore processes 4 batch rows per iteration.
    const float4* tbl4  = (const float4*)tbl;
    const int     q     = tid & 7;   // float4 slot within the 32-float embedding
    const int     group = tid >> 3;  // which of the 32 batch rows this block-iter covers

#pragma unroll
    for (int iter = 0; iter < CHUNK / 32; ++iter) {
        const int b = base + iter * 32 + group;
        // 20 leaf ids for (b, s): 80 contiguous, 16B-aligned bytes -> 5 x int4
        const int4* lp = (const int4*)(leaves + (size_t)b * N_TREES + s * M_PER_SPLIT);

        float ax = 0.f, ay = 0.f, az = 0.f, aw = 0.f;
#pragma unroll
        for (int k = 0; k < 5; ++k) {
            const int4 lv = lp[k];
            const int  t0 = 4 * k;
            // row = t*MAXLEAF + leaf; each row = 8 float4s; ds_load_b128 per gather
            const float4 v0 = tbl4[((t0 + 0) * MAXLEAF + lv.x) * 8 + q];
            const float4 v1 = tbl4[((t0 + 1) * MAXLEAF + lv.y) * 8 + q];
            const float4 v2 = tbl4[((t0 + 2) * MAXLEAF + lv.z) * 8 + q];
            const float4 v3 = tbl4[((t0 + 3) * MAXLEAF + lv.w) * 8 + q];
            ax += v0.x + v1.x + v2.x + v3.x;
            ay += v0.y + v1.y + v2.y + v3.y;
            az += v0.z + v1.z + v2.z + v3.z;
            aw += v0.w + v1.w + v2.w + v3.w;
        }

        float4 r; r.x = ax; r.y = ay; r.z = az; r.w = aw;
        // out[b, s*32 + q*4 .. +3], 16B-aligned, coalesced 128B per batch row
        ((float4*)(out + (size_t)b * (N_SPLIT * EMBSIZE) + s * EMBSIZE))[q] = r;
    }
}

extern "C" void kernel_launch(void* const* d_in, const int* in_sizes, int n_in,
                              void* d_out, int out_size, void* d_ws, size_t ws_size,
                              hipStream_t stream) {
    const int*   leaves  = (const int*)d_in[0];    // [4096, 2000] int32
    const float* embed_w = (const float*)d_in[1];  // [100, 1280, 32] f32
    float*       out     = (float*)d_out;          // [4096, 3200] f32

    dim3 grid(N_SPLIT, BATCH / CHUNK);             // (100, 32)
    Leaf2Embedding_gather_kernel<<<grid, THREADS, 0, stream>>>(leaves, embed_w, out);
}